// compute_contra_loss_86002425135457
// MI455X (gfx1250) — compile-verified
//
#include <hip/hip_runtime.h>
#include <hip/hip_bf16.h>
#include <math.h>

typedef float v2f __attribute__((ext_vector_type(2)));
typedef float v8f __attribute__((ext_vector_type(8)));

#define EPS_F     1e-6f
#define MARGIN_F  20.0f
#define CLAMP2_F  100.0f     // min(self_dist,10)^2 == min(ssd,100)
#define DIM       512
#define INF_BITS  0x7f800000u

// ---------------------------------------------------------------------------
// Kernel 1: per-row norms + self-distance + init min-d2 buffer.
// One 256-thread block per row (8 wave32).
// ---------------------------------------------------------------------------
__global__ __launch_bounds__(256) void row_stats_kernel(
    const float* __restrict__ x, const float* __restrict__ z,
    float* __restrict__ sq_x, float* __restrict__ sq_a,
    float* __restrict__ ssd, unsigned int* __restrict__ minb, int D)
{
    const int row  = blockIdx.x;
    const int tid  = threadIdx.x;
    const int lane = tid & 31;
    const int wv   = tid >> 5;

    const float* xr = x + (size_t)row * D;
    const float* zr = z + (size_t)row * D;

    float sxx = 0.f, saa = 0.f, sdd = 0.f;
    for (int c = tid; c < D; c += 256) {
        float xv = xr[c];
        float zv = zr[c];
        float av = xv + EPS_F;
        float dv = xv - zv + EPS_F;
        sxx += xv * xv;
        saa += av * av;
        sdd += dv * dv;
    }
    #pragma unroll
    for (int m = 16; m >= 1; m >>= 1) {
        sxx += __shfl_xor(sxx, m, 32);
        saa += __shfl_xor(saa, m, 32);
        sdd += __shfl_xor(sdd, m, 32);
    }
    __shared__ float s0[8], s1[8], s2[8];
    if (lane == 0) { s0[wv] = sxx; s1[wv] = saa; s2[wv] = sdd; }
    __syncthreads();
    if (tid == 0) {
        float a = 0.f, b = 0.f, c = 0.f;
        #pragma unroll
        for (int i = 0; i < 8; ++i) { a += s0[i]; b += s1[i]; c += s2[i]; }
        sq_x[row] = a;
        sq_a[row] = b;
        ssd[row]  = c;
        minb[row] = INF_BITS;
    }
}

// ---------------------------------------------------------------------------
// Kernel 2: WMMA fp32 GEMM + running min of off-diagonal squared distance.
// Block = 256 threads (8 wave32); block tile 256 rows x 128 cols.
// Each wave: 2 M-tiles x 8 N-tiles (32 rows x 128 cols), double-buffered K.
// ---------------------------------------------------------------------------
__global__ __launch_bounds__(256) void min_dist_wmma_kernel(
    const float* __restrict__ x,
    const float* __restrict__ sq_x, const float* __restrict__ sq_a,
    unsigned int* __restrict__ minb, int N, int D)
{
    const int lane    = threadIdx.x & 31;
    const int wave    = threadIdx.x >> 5;
    const int rowBase = blockIdx.y * 256 + wave * 32;   // 2 M-tiles of 16
    const int colBase = blockIdx.x * 128;               // 8 N-tiles of 16
    const int mrow    = lane & 15;
    const int half    = lane >> 4;                       // K sub-pair selector
    const int kHalf   = half * 2;

    const float* aBase = x + (size_t)(rowBase + mrow) * DIM + kHalf;
    const float* bBase = x + (size_t)(colBase + mrow) * DIM + kHalf;

    v8f acc[2][8];
    #pragma unroll
    for (int m = 0; m < 2; ++m)
        #pragma unroll
        for (int t = 0; t < 8; ++t) acc[m][t] = (v8f)0.0f;

    v2f a0[2], a1[2], b0[8], b1[8];

    // Preload K-step 0 into buffer 0.
    #pragma unroll
    for (int m = 0; m < 2; ++m)
        a0[m] = *reinterpret_cast<const v2f*>(aBase + m * 16 * DIM) + EPS_F;
    #pragma unroll
    for (int t = 0; t < 8; ++t)
        b0[t] = *reinterpret_cast<const v2f*>(bBase + t * 16 * DIM);

    for (int k0 = 0; k0 < DIM; k0 += 8) {
        const int k1 = k0 + 4;
        // Prefetch K-step k1 into buffer 1 (independent registers -> no WAR).
        #pragma unroll
        for (int m = 0; m < 2; ++m)
            a1[m] = *reinterpret_cast<const v2f*>(aBase + m * 16 * DIM + k1) + EPS_F;
        #pragma unroll
        for (int t = 0; t < 8; ++t)
            b1[t] = *reinterpret_cast<const v2f*>(bBase + t * 16 * DIM + k1);

        // 16 back-to-back WMMAs on buffer 0.
        #pragma unroll
        for (int m = 0; m < 2; ++m)
            #pragma unroll
            for (int t = 0; t < 8; ++t)
                acc[m][t] = __builtin_amdgcn_wmma_f32_16x16x4_f32(
                    false, a0[m], false, b0[t], (short)0, acc[m][t], false, false);

        // Prefetch K-step k0+8 into buffer 0 (wrap on last iter, stays in bounds).
        const int k2 = (k0 + 8) < DIM ? (k0 + 8) : 0;
        #pragma unroll
        for (int m = 0; m < 2; ++m)
            a0[m] = *reinterpret_cast<const v2f*>(aBase + m * 16 * DIM + k2) + EPS_F;
        #pragma unroll
        for (int t = 0; t < 8; ++t)
            b0[t] = *reinterpret_cast<const v2f*>(bBase + t * 16 * DIM + k2);

        // 16 back-to-back WMMAs on buffer 1.
        #pragma unroll
        for (int m = 0; m < 2; ++m)
            #pragma unroll
            for (int t = 0; t < 8; ++t)
                acc[m][t] = __builtin_amdgcn_wmma_f32_16x16x4_f32(
                    false, a1[m], false, b1[t], (short)0, acc[m][t], false, false);
    }

    // Epilogue: C/D layout => lane holds N = lane&15, rows M = r + 8*half.
    const float inf = __uint_as_float(INF_BITS);
    float sa[2][8];
    #pragma unroll
    for (int m = 0; m < 2; ++m)
        #pragma unroll
        for (int r = 0; r < 8; ++r)
            sa[m][r] = sq_a[rowBase + m * 16 + r + 8 * half];

    float rmin[2][8];
    #pragma unroll
    for (int m = 0; m < 2; ++m)
        #pragma unroll
        for (int r = 0; r < 8; ++r) rmin[m][r] = inf;

    #pragma unroll
    for (int t = 0; t < 8; ++t) {
        const int col = colBase + t * 16 + mrow;
        const float sx = sq_x[col];
        #pragma unroll
        for (int m = 0; m < 2; ++m) {
            #pragma unroll
            for (int r = 0; r < 8; ++r) {
                const int row = rowBase + m * 16 + r + 8 * half;
                float d2 = sa[m][r] + sx - 2.0f * acc[m][t][r];
                d2 = fmaxf(d2, 0.0f);
                if (row == col) d2 = inf;               // exclude i == j
                rmin[m][r] = fminf(rmin[m][r], d2);
            }
        }
    }

    // Min across the 16 lanes sharing each row (masks < 16 keep halves apart).
    #pragma unroll
    for (int msk = 1; msk <= 8; msk <<= 1) {
        #pragma unroll
        for (int m = 0; m < 2; ++m)
            #pragma unroll
            for (int r = 0; r < 8; ++r)
                rmin[m][r] = fminf(rmin[m][r], __shfl_xor(rmin[m][r], msk, 32));
    }

    if (mrow == 0) {     // lanes 0 and 16 publish 8 rows per M-tile each
        #pragma unroll
        for (int m = 0; m < 2; ++m)
            #pragma unroll
            for (int r = 0; r < 8; ++r) {
                const int row = rowBase + m * 16 + r + 8 * half;
                atomicMin(&minb[row], __float_as_uint(rmin[m][r]));
            }
    }
}

// ---------------------------------------------------------------------------
// Kernel 3: loss = mean_i [ min(ssd_i, 100) + max(20 - sqrt(min_d2_i), 0)^2 ]
// ---------------------------------------------------------------------------
__global__ __launch_bounds__(256) void finalize_kernel(
    const float* __restrict__ ssd, const unsigned int* __restrict__ minb,
    float* __restrict__ out, int N)
{
    const int tid  = threadIdx.x;
    const int lane = tid & 31;
    const int wv   = tid >> 5;

    float acc = 0.f;
    for (int i = tid; i < N; i += 256) {
        float selfsq = fminf(ssd[i], CLAMP2_F);
        float d2     = __uint_as_float(minb[i]);
        float mind   = sqrtf(fmaxf(d2, 0.0f));
        float hinge  = fmaxf(MARGIN_F - mind, 0.0f);
        acc += selfsq + hinge * hinge;
    }
    #pragma unroll
    for (int m = 16; m >= 1; m >>= 1) acc += __shfl_xor(acc, m, 32);

    __shared__ float s[8];
    if (lane == 0) s[wv] = acc;
    __syncthreads();
    if (tid == 0) {
        float t = 0.f;
        #pragma unroll
        for (int i = 0; i < 8; ++i) t += s[i];
        out[0] = t / (float)N;
    }
}

// ---------------------------------------------------------------------------
extern "C" void kernel_launch(void* const* d_in, const int* in_sizes, int n_in,
                              void* d_out, int out_size, void* d_ws, size_t ws_size,
                              hipStream_t stream) {
    (void)n_in; (void)out_size; (void)ws_size;
    const float* x = (const float*)d_in[0];
    const float* z = (const float*)d_in[1];
    const int D = DIM;
    const int N = in_sizes[0] / D;          // 8192

    // Workspace layout (floats): sq_x | sq_a | ssd | minbits   (4*N*4 B = 128 KB)
    float* ws          = (float*)d_ws;
    float* sq_x        = ws;
    float* sq_a        = ws + N;
    float* ssd         = ws + 2 * N;
    unsigned int* minb = (unsigned int*)(ws + 3 * N);

    row_stats_kernel<<<N, 256, 0, stream>>>(x, z, sq_x, sq_a, ssd, minb, D);

    dim3 grid(N / 128, N / 256);            // 64 x 32 = 2048 workgroups
    min_dist_wmma_kernel<<<grid, 256, 0, stream>>>(x, sq_x, sq_a, minb, N, D);

    finalize_kernel<<<1, 256, 0, stream>>>(ssd, minb, (float*)d_out, N);
}